// NetVLAD_23106924053262
// MI455X (gfx1250) — compile-verified
//
#include <hip/hip_runtime.h>

typedef __attribute__((ext_vector_type(2))) float v2f;
typedef __attribute__((ext_vector_type(4))) float v4f;
typedef __attribute__((ext_vector_type(8))) float v8f;
typedef __attribute__((ext_vector_type(4))) int   v4i;

#define N_CFG 32
#define C_CFG 128
#define K_CFG 64
#define S_CFG 12544            // 112*112
#define CHUNK 128              // pixels per chunk
#define CHUNKS_PER_BLOCK 7
#define BLOCKS_PER_N 14        // 14*7*128 = 12544 exactly

// LDS layout (float offsets), total 49152 floats = 192 KB (double-buffered x)
#define LDS_CW 0               // conv_w: 64*128            (8192)
#define LDS_X0 8192            // x tile buf0: 128C x 128s  (16384)
#define LDS_X1 24576           // x tile buf1: 128C x 128s  (16384)
#define LDS_A  40960           // a tile: 64K x 128s        (8192)
#define LDS_FLOATS 49152

#if defined(__AMDGCN__) && \
    __has_builtin(__builtin_amdgcn_global_load_async_to_lds_b128) && \
    __has_builtin(__builtin_amdgcn_s_wait_asynccnt)
#define HAS_ASYNC_LDS 1
#else
#define HAS_ASYNC_LDS 0
#endif

__global__ void netvlad_zero(float* __restrict__ ws, int n) {
  int i = blockIdx.x * blockDim.x + threadIdx.x;
  if (i < n) ws[i] = 0.0f;
}

#if HAS_ASYNC_LDS
// Issue this wave's 16 rows of the x chunk as async DMA global->LDS (b128/lane).
__device__ __forceinline__ void issue_chunk_async(const float* __restrict__ xb_n,
                                                  int s0, float* smem, int xoff,
                                                  int wave, int lane) {
  #pragma unroll
  for (int r = 0; r < 16; ++r) {
    const int c = wave * 16 + r;
    const float* gp = xb_n + (size_t)c * S_CFG + s0 + lane * 4;
    __builtin_amdgcn_global_load_async_to_lds_b128(
        (__attribute__((address_space(1))) v4i*)gp,
        (__attribute__((address_space(3))) v4i*)&smem[xoff + c * CHUNK + lane * 4],
        0, 0);
  }
}
#endif

__global__ void __launch_bounds__(256) netvlad_main(
    const float* __restrict__ x,        // (N, C, S)
    const float* __restrict__ conv_w,   // (K, C)
    float* __restrict__ vlad_acc,       // (N, K, C) accumulators
    float* __restrict__ asum_acc) {     // (N, K)
  extern __shared__ float smem[];
  const int tid  = threadIdx.x;
  const int wave = tid >> 5;
  const int lane = tid & 31;
  const int l16  = lane & 15;
  const int hi   = (lane >> 4) & 1;       // which half-wave
  const int n    = blockIdx.y;
  const float* xb_n = x + (size_t)n * C_CFG * S_CFG;
  const int sblk = blockIdx.x * CHUNKS_PER_BLOCK * CHUNK;

  // Stage conv_w (64x128 f32 = 32 KB) into LDS once.
  {
    const v4f* src = (const v4f*)conv_w;
    v4f* dst = (v4f*)&smem[LDS_CW];
    for (int i = tid; i < (K_CFG * C_CFG) / 4; i += 256) dst[i] = src[i];
  }

  const v8f zero8 = {0.f, 0.f, 0.f, 0.f, 0.f, 0.f, 0.f, 0.f};
  v8f vacc[4] = {zero8, zero8, zero8, zero8};  // persistent GEMM2 accum: 64K x 16C slice
  float asum_reg = 0.0f;                       // threads 0..63 hold asum for k = tid

#if HAS_ASYNC_LDS
  issue_chunk_async(xb_n, sblk, smem, LDS_X0, wave, lane);  // prime buffer 0
#endif

  for (int ch = 0; ch < CHUNKS_PER_BLOCK; ++ch) {
    const int cur   = ch & 1;
    const int xbase = cur ? LDS_X1 : LDS_X0;

#if HAS_ASYNC_LDS
    __builtin_amdgcn_s_wait_asynccnt(0);  // own chunk-ch DMA landed in LDS
    __syncthreads();                      // all waves landed; prev readers of other buf done
    if (ch + 1 < CHUNKS_PER_BLOCK)        // overlap next chunk DMA with compute below
      issue_chunk_async(xb_n, sblk + (ch + 1) * CHUNK, smem,
                        cur ? LDS_X0 : LDS_X1, wave, lane);
#else
    __syncthreads();  // protect LDS tile from previous-iteration readers
    {
      const float* xb = xb_n + sblk + ch * CHUNK;
      #pragma unroll
      for (int r = 0; r < 16; ++r) {
        const int c = wave * 16 + r;
        const float* rowp = xb + (size_t)c * S_CFG + lane * 4;
        v4f v = *(const v4f*)rowp;
        __builtin_prefetch(rowp + CHUNK, 0, 0);
        *(v4f*)&smem[xbase + c * CHUNK + lane * 4] = v;
      }
    }
    __syncthreads();
#endif

    // ---- Phase 2: GEMM1 (logits = conv_w @ x) + softmax, strip = wave ----
    {
      const int cl = wave * 16 + l16;  // local pixel column owned by this lane
      v8f acc[4] = {zero8, zero8, zero8, zero8};
      #pragma unroll
      for (int kc = 0; kc < 32; ++kc) {            // reduction over C in steps of 4
        const int r0 = kc * 4 + 2 * hi;
        v2f bf;
        bf.x = smem[xbase + (r0 + 0) * CHUNK + cl];
        bf.y = smem[xbase + (r0 + 1) * CHUNK + cl];
        #pragma unroll
        for (int mt = 0; mt < 4; ++mt) {           // 4 K-tiles of 16
          v2f af = *(const v2f*)&smem[LDS_CW + (mt * 16 + l16) * C_CFG + r0];
          acc[mt] = __builtin_amdgcn_wmma_f32_16x16x4_f32(
              false, af, false, bf, (short)0, acc[mt], false, false);
        }
      }
      // Softmax over K=64: values live in acc[4][8] here + the lane^16 partner.
      float m = -3.0e38f;
      #pragma unroll
      for (int mt = 0; mt < 4; ++mt)
        #pragma unroll
        for (int j = 0; j < 8; ++j) m = fmaxf(m, acc[mt][j]);
      m = fmaxf(m, __shfl_xor(m, 16, 32));
      float ssum = 0.0f;
      #pragma unroll
      for (int mt = 0; mt < 4; ++mt)
        #pragma unroll
        for (int j = 0; j < 8; ++j) {
          float e = __expf(acc[mt][j] - m);
          acc[mt][j] = e;
          ssum += e;
        }
      ssum += __shfl_xor(ssum, 16, 32);
      const float inv = 1.0f / ssum;
      #pragma unroll
      for (int mt = 0; mt < 4; ++mt)
        #pragma unroll
        for (int j = 0; j < 8; ++j)
          smem[LDS_A + (mt * 16 + j + 8 * hi) * CHUNK + cl] = acc[mt][j] * inv;
    }
    __syncthreads();

    // ---- Phase 2.5: asum partial (waves 0-1, uniform per-wave) ----
    if (tid < K_CFG) {
      const float* ar = &smem[LDS_A + tid * CHUNK];
      float s = 0.0f;
      #pragma unroll
      for (int i = 0; i < CHUNK; i += 4) {
        v4f v = *(const v4f*)(ar + i);
        s += v.x + v.y + v.z + v.w;
      }
      asum_reg += s;
    }

    // ---- Phase 3: GEMM2 (vlad += a @ x^T), this wave owns C cols [c0,c0+16) ----
    {
      const int c0 = wave * 16;
      for (int strip = 0; strip < 8; ++strip) {
        #pragma unroll
        for (int q = 0; q < 4; ++q) {
          const int sb = strip * 16 + q * 4 + 2 * hi;  // reduction (pixel) index base
          v2f bf = *(const v2f*)&smem[xbase + (c0 + l16) * CHUNK + sb];
          #pragma unroll
          for (int mt = 0; mt < 4; ++mt) {
            v2f af = *(const v2f*)&smem[LDS_A + (mt * 16 + l16) * CHUNK + sb];
            vacc[mt] = __builtin_amdgcn_wmma_f32_16x16x4_f32(
                false, af, false, bf, (short)0, vacc[mt], false, false);
          }
        }
      }
    }
  }

  // ---- Epilogue: accumulate partial VLAD + asum to global ----
  {
    float* base = vlad_acc + (size_t)n * K_CFG * C_CFG;
    const int c = wave * 16 + l16;
    #pragma unroll
    for (int mt = 0; mt < 4; ++mt)
      #pragma unroll
      for (int j = 0; j < 8; ++j) {
        const int M = mt * 16 + j + 8 * hi;
        atomicAdd(base + M * C_CFG + c, vacc[mt][j]);
      }
  }
  if (tid < K_CFG) atomicAdd(asum_acc + n * K_CFG + tid, asum_reg);
}

// Per (n,k): subtract asum*centroid, intra-normalize over C, emit + global ss.
__global__ void __launch_bounds__(128) netvlad_finalize1(
    const float* __restrict__ vlad_acc, const float* __restrict__ asum_acc,
    const float* __restrict__ cent, float* __restrict__ out,
    float* __restrict__ gnorm) {
  const int k = blockIdx.x, n = blockIdx.y, c = threadIdx.x;
  const float v = vlad_acc[((size_t)n * K_CFG + k) * C_CFG + c] -
                  asum_acc[n * K_CFG + k] * cent[k * C_CFG + c];
  float ss = v * v;
  #pragma unroll
  for (int o = 16; o > 0; o >>= 1) ss += __shfl_xor(ss, o, 32);
  __shared__ float red[4];
  if ((threadIdx.x & 31) == 0) red[threadIdx.x >> 5] = ss;
  __syncthreads();
  ss = red[0] + red[1] + red[2] + red[3];
  const float norm = fmaxf(sqrtf(ss), 1e-12f);
  out[(size_t)n * (K_CFG * C_CFG) + k * C_CFG + c] = v / norm;
  if (threadIdx.x == 0) atomicAdd(&gnorm[n], ss / (norm * norm));
}

__global__ void netvlad_finalize2(float* __restrict__ out,
                                  const float* __restrict__ gnorm) {
  const int n = blockIdx.x;
  const float inv = 1.0f / fmaxf(sqrtf(gnorm[n]), 1e-12f);
  for (int i = threadIdx.x; i < K_CFG * C_CFG; i += blockDim.x)
    out[(size_t)n * (K_CFG * C_CFG) + i] *= inv;
}

extern "C" void kernel_launch(void* const* d_in, const int* in_sizes, int n_in,
                              void* d_out, int out_size, void* d_ws, size_t ws_size,
                              hipStream_t stream) {
  const float* x    = (const float*)d_in[0];  // (N,C,H,W) f32
  const float* cw   = (const float*)d_in[1];  // (K,C) f32
  const float* cent = (const float*)d_in[2];  // (K,C) f32
  float* out = (float*)d_out;                 // (N, K*C) f32

  float* vlad_acc = (float*)d_ws;                              // 262144 floats
  float* asum_acc = vlad_acc + (size_t)N_CFG * K_CFG * C_CFG;  // 2048 floats
  float* gnorm    = asum_acc + (size_t)N_CFG * K_CFG;          // 32 floats
  const int nz = N_CFG * K_CFG * C_CFG + N_CFG * K_CFG + N_CFG;

  netvlad_zero<<<(nz + 255) / 256, 256, 0, stream>>>(vlad_acc, nz);

  dim3 grid(BLOCKS_PER_N, N_CFG);
  netvlad_main<<<grid, 256, LDS_FLOATS * sizeof(float), stream>>>(
      x, cw, vlad_acc, asum_acc);

  netvlad_finalize1<<<dim3(K_CFG, N_CFG), 128, 0, stream>>>(
      vlad_acc, asum_acc, cent, out, gnorm);
  netvlad_finalize2<<<N_CFG, 256, 0, stream>>>(out, gnorm);
}